// HeteroGCN_33303176413370
// MI455X (gfx1250) — compile-verified
//
#include <hip/hip_runtime.h>
#include <hip/hip_bf16.h>

// ---------------------------------------------------------------------------
// HeteroGCN: h = GCNConv(x, edge_index; W_gcn, b_gcn)  (sym-norm, self loops)
//            out[l] = concat(h[eli0[l]], h[eli1[l]]) @ W_lin + b_lin
// HID = 128, OUT = 5.
// ---------------------------------------------------------------------------

#define HID 128
#define OUT_DIM 5

typedef float v2f __attribute__((ext_vector_type(2)));
typedef float v8f __attribute__((ext_vector_type(8)));

// --- degree: deg[i] = 1 (self loop) --------------------------------------
__global__ void deg_init_kernel(float* __restrict__ deg, int n) {
    int i = blockIdx.x * blockDim.x + threadIdx.x;
    if (i < n) deg[i] = 1.0f;
}

// --- degree: += 1 per incoming edge --------------------------------------
__global__ void deg_accum_kernel(const int* __restrict__ dst, float* __restrict__ deg, int e) {
    int i = blockIdx.x * blockDim.x + threadIdx.x;
    if (i < e) atomicAdd(&deg[dst[i]], 1.0f);
}

// --- deg -> d^{-1/2} in place --------------------------------------------
__global__ void deg_rsqrt_kernel(float* __restrict__ deg, int n) {
    int i = blockIdx.x * blockDim.x + threadIdx.x;
    if (i < n) {
        float d = deg[i];
        deg[i] = (d > 0.0f) ? rsqrtf(d) : 0.0f;
    }
}

// --- xw = x @ W  via V_WMMA_F32_16X16X4_F32 -------------------------------
// One block = 16 output rows; 8 waves each own one 16-wide column tile.
// A strip (16 x 128) staged in LDS; W (64 KB, shared by all blocks) streams
// through L2 / WGP$.
__global__ __launch_bounds__(256) void gemm_xw_wmma_kernel(
        const float* __restrict__ x, const float* __restrict__ W,
        float* __restrict__ xw, int nrows) {
    __shared__ float As[16][HID + 4];  // +4 pad: dodge LDS bank conflicts

    const int tid  = threadIdx.x;
    const int row0 = blockIdx.x * 16;

    // Cooperative strip load: 2048 floats = 512 float4, 256 threads x 2.
    for (int i = tid; i < 512; i += 256) {
        int r  = i >> 5;       // 32 float4 per row
        int c4 = i & 31;
        float4 v = make_float4(0.f, 0.f, 0.f, 0.f);
        int gr = row0 + r;
        if (gr < nrows) v = ((const float4*)(x + (size_t)gr * HID))[c4];
        As[r][c4 * 4 + 0] = v.x;
        As[r][c4 * 4 + 1] = v.y;
        As[r][c4 * 4 + 2] = v.z;
        As[r][c4 * 4 + 3] = v.w;
    }
    __syncthreads();

    const int wave = tid >> 5;            // 0..7 -> N tile
    const int lane = tid & 31;
    const int m    = lane & 15;           // A row
    const int kk   = (lane >> 4) * 2;     // lanes 16-31 carry K+2,K+3
    const int ncol = wave * 16 + (lane & 15);

    v8f c = {};
    for (int k = 0; k < HID; k += 4) {
        v2f a, b;
        // A 16x4 f32 layout: lanes 0-15 {K,K+1}, lanes 16-31 {K+2,K+3}
        a.x = As[m][k + kk];
        a.y = As[m][k + kk + 1];
        // B 4x16 f32 layout (mirror of A): lanes 0-15 rows {K,K+1}, 16-31 {K+2,K+3}
        b.x = W[(size_t)(k + kk) * HID + ncol];
        b.y = W[(size_t)(k + kk + 1) * HID + ncol];
        c = __builtin_amdgcn_wmma_f32_16x16x4_f32(
                /*neg_a=*/false, a, /*neg_b=*/false, b,
                /*c_mod=*/(short)0, c, /*reuse_a=*/false, /*reuse_b=*/false);
    }

    // C/D 16x16 f32: VGPR g -> M = g + (lane>=16 ? 8 : 0), N = lane & 15.
    const int mbase = (lane >> 4) * 8;
#pragma unroll
    for (int g = 0; g < 8; ++g) {
        int gr = row0 + mbase + g;
        if (gr < nrows) xw[(size_t)gr * HID + ncol] = c[g];
    }
}

// --- h = b_gcn + dinv^2 * xw  (self-loop term) -----------------------------
// One wave per node; lane owns 4 consecutive columns (float4).
__global__ void h_init_kernel(const float* __restrict__ xw,
                              const float* __restrict__ dinv,
                              const float* __restrict__ bias,
                              float* __restrict__ h, int n) {
    int t = blockIdx.x * blockDim.x + threadIdx.x;
    int node = t >> 5;
    int c4   = t & 31;
    if (node >= n) return;
    float di   = dinv[node];
    float norm = di * di;
    float4 xv = ((const float4*)(xw + (size_t)node * HID))[c4];
    float4 bv = ((const float4*)bias)[c4];
    float4 o  = make_float4(bv.x + norm * xv.x, bv.y + norm * xv.y,
                            bv.z + norm * xv.z, bv.w + norm * xv.w);
    ((float4*)(h + (size_t)node * HID))[c4] = o;
}

// --- h[dst] += norm(e) * xw[src]  (the bandwidth-dominant phase) -----------
// One wave per edge, lane = 4-column group: coalesced float4 gather of
// xw[src], then 4 non-returning f32 atomics into h[dst] (resolved in L2;
// h is 51 MB and stays resident in the 192 MB L2).
__global__ void scatter_edges_kernel(const int* __restrict__ src,
                                     const int* __restrict__ dst,
                                     const float* __restrict__ xw,
                                     const float* __restrict__ dinv,
                                     float* __restrict__ h, int e) {
    int t = blockIdx.x * blockDim.x + threadIdx.x;
    int edge = t >> 5;
    int c4   = t & 31;
    if (edge >= e) return;
    int s = src[edge];
    int d = dst[edge];
    float norm = dinv[s] * dinv[d];
    float4 xv = ((const float4*)(xw + (size_t)s * HID))[c4];
    float* hp = h + (size_t)d * HID + c4 * 4;
    atomicAdd(hp + 0, norm * xv.x);
    atomicAdd(hp + 1, norm * xv.y);
    atomicAdd(hp + 2, norm * xv.z);
    atomicAdd(hp + 3, norm * xv.w);
}

// --- out[l] = concat(h[a], h[b]) @ W_lin + b_lin ---------------------------
// One wave per label edge; lane covers 4 k of each half (2*128 = 256 = 32*8).
// W_lin (256x5 row-major) staged in LDS; wave32 shuffle reduction.
__global__ __launch_bounds__(256) void edge_mlp_kernel(
        const int* __restrict__ ia, const int* __restrict__ ib,
        const float* __restrict__ h, const float* __restrict__ W,
        const float* __restrict__ bias, float* __restrict__ out, int nl) {
    __shared__ float Ws[2 * HID * OUT_DIM];  // 1280 floats = 5 KB
    int tid = threadIdx.x;
    for (int i = tid; i < 2 * HID * OUT_DIM; i += 256) Ws[i] = W[i];
    __syncthreads();

    int wave = tid >> 5;
    int lane = tid & 31;
    int l = blockIdx.x * 8 + wave;
    if (l >= nl) return;

    int na = ia[l];
    int nb = ib[l];
    float4 va = ((const float4*)(h + (size_t)na * HID))[lane];  // k = lane*4+j
    float4 vb = ((const float4*)(h + (size_t)nb * HID))[lane];  // k = 128+lane*4+j
    float av[4] = {va.x, va.y, va.z, va.w};
    float bv[4] = {vb.x, vb.y, vb.z, vb.w};

    float acc[OUT_DIM] = {0.f, 0.f, 0.f, 0.f, 0.f};
#pragma unroll
    for (int j = 0; j < 4; ++j) {
        int ka = lane * 4 + j;
        int kb = HID + lane * 4 + j;
#pragma unroll
        for (int o = 0; o < OUT_DIM; ++o)
            acc[o] += av[j] * Ws[ka * OUT_DIM + o] + bv[j] * Ws[kb * OUT_DIM + o];
    }
#pragma unroll
    for (int off = 16; off > 0; off >>= 1)
#pragma unroll
        for (int o = 0; o < OUT_DIM; ++o)
            acc[o] += __shfl_down(acc[o], off, 32);

    if (lane == 0) {
#pragma unroll
        for (int o = 0; o < OUT_DIM; ++o)
            out[(size_t)l * OUT_DIM + o] = acc[o] + bias[o];
    }
}

// ---------------------------------------------------------------------------
extern "C" void kernel_launch(void* const* d_in, const int* in_sizes, int n_in,
                              void* d_out, int out_size, void* d_ws, size_t ws_size,
                              hipStream_t stream) {
    const float* x     = (const float*)d_in[0];
    const int*   ei    = (const int*)d_in[1];   // [2, E] src row then dst row
    const int*   eli   = (const int*)d_in[2];   // [2, L]
    const float* W_gcn = (const float*)d_in[3];
    const float* b_gcn = (const float*)d_in[4];
    const float* W_lin = (const float*)d_in[5];
    const float* b_lin = (const float*)d_in[6];

    const int N = in_sizes[0] / HID;
    const int E = in_sizes[1] / 2;
    const int L = in_sizes[2] / 2;

    const int* e_src = ei;
    const int* e_dst = ei + E;
    const int* l_a   = eli;
    const int* l_b   = eli + L;

    // workspace layout: xw [N*128] | h [N*128] | deg/dinv [N]
    float* xw  = (float*)d_ws;
    float* h   = xw + (size_t)N * HID;
    float* deg = h + (size_t)N * HID;

    float* out = (float*)d_out;

    const int B = 256;

    // 1) degree with self loops
    deg_init_kernel<<<(N + B - 1) / B, B, 0, stream>>>(deg, N);
    deg_accum_kernel<<<(E + B - 1) / B, B, 0, stream>>>(e_dst, deg, E);
    deg_rsqrt_kernel<<<(N + B - 1) / B, B, 0, stream>>>(deg, N);

    // 2) xw = x @ W_gcn  (WMMA)
    gemm_xw_wmma_kernel<<<(N + 15) / 16, 256, 0, stream>>>(x, W_gcn, xw, N);

    // 3) h = b + dinv^2 * xw  (self loop), then edge scatter-add
    {
        long long threads = (long long)N * 32;
        h_init_kernel<<<(unsigned)((threads + B - 1) / B), B, 0, stream>>>(xw, deg, b_gcn, h, N);
    }
    {
        long long threads = (long long)E * 32;
        scatter_edges_kernel<<<(unsigned)((threads + B - 1) / B), B, 0, stream>>>(
            e_src, e_dst, xw, deg, h, E);
    }

    // 4) link-prediction head
    edge_mlp_kernel<<<(L + 7) / 8, 256, 0, stream>>>(l_a, l_b, h, W_lin, b_lin, out, L);

    (void)n_in; (void)out_size; (void)ws_size;
}